// ConvLayer_49959059587609
// MI455X (gfx1250) — compile-verified
//
#include <hip/hip_runtime.h>
#include <float.h>
#include <math.h>

#define BSZ 4
#define CC  4
#define NN  2048
#define DD  16
#define KK  32
#define MM  64
#define FTOT 256   /* K*C (theta) + 2*C*D (gated features) = 128 + 128 */

typedef float v2f __attribute__((ext_vector_type(2)));
typedef float v8f __attribute__((ext_vector_type(8)));
typedef unsigned int u32;
typedef u32 u32x4 __attribute__((ext_vector_type(4)));
typedef int  i32x4 __attribute__((ext_vector_type(4)));
typedef int  i32x8 __attribute__((ext_vector_type(8)));

// ---------------------------------------------------------------------------
// Kernel C: pack combined weight  W[f][m], f<128 -> angle_weight[m][f] (transpose),
//           f>=128 -> scalar_weight[f-128][m]
// ---------------------------------------------------------------------------
__global__ __launch_bounds__(256) void build_w_kernel(const float* __restrict__ aw,
                                                      const float* __restrict__ sw,
                                                      float* __restrict__ W) {
    int t = blockIdx.x * 256 + threadIdx.x;        // 0 .. 256*64-1
    int f = t >> 6;
    int m = t & 63;
    W[t] = (f < 128) ? aw[m * 128 + f] : sw[(f - 128) * 64 + m];
}

// ---------------------------------------------------------------------------
// Kernel A: per (b,c,n) block: TDM-staged pos tile, register-resident top-33
// selection (shfl_xor butterflies, 2 barriers/iter), direction/cosine features,
// radius gate, k-summed gated features. Writes feature row X[(b*N+n)*256 + ...].
// ---------------------------------------------------------------------------
__global__ __launch_bounds__(256) void knn_gate_kernel(
    const float* __restrict__ pos, const float* __restrict__ fea,
    const float* __restrict__ mask, const float* __restrict__ rw1,
    const float* __restrict__ rw2, float* __restrict__ X) {

    __shared__ float s_pos[NN * 3];    // 24 KB, filled by TDM
    __shared__ float s_val[8];         // per-wave argmin value
    __shared__ int   s_idx[8];         // per-wave argmin index
    __shared__ int   s_sel[KK + 1];
    __shared__ float s_seld[KK + 1];
    __shared__ float s_gf[KK * DD];    // gate * neighbor-features
    __shared__ float s_gate[KK];

    const int n = blockIdx.x, c = blockIdx.y, b = blockIdx.z;
    const int tid  = threadIdx.x;
    const int lane = tid & 31;
    const int w    = tid >> 5;
    const float* posc = pos + (size_t)(b * CC + c) * NN * 3;
    const float* feac = fea + (size_t)(b * CC + c) * NN * DD;

    // ---- stage pos tile (2048x3 f32 = 24 KB) into LDS via the Tensor Data Mover
#if __has_builtin(__builtin_amdgcn_tensor_load_to_lds)
    if (tid < 32) {                                   // one wave issues the DMA
        const u32 lds_off = (u32)(uintptr_t)(void*)s_pos;   // LDS byte address of s_pos
        const unsigned long long ga = (unsigned long long)(uintptr_t)posc;
        u32x4 g0;
        g0.x = 1u;                                    // count=1 (valid), is_restore=0
        g0.y = lds_off;                               // lds_addr (bytes)
        g0.z = (u32)(ga & 0xffffffffull);             // global_addr[31:0]
        g0.w = (u32)((ga >> 32) & 0x1ffffffull)       // global_addr[56:32]
             | (2u << 30);                            // type = 2 ("image")
        i32x8 g1;
        g1[0] = (int)(2u << 16);                      // data_size = 4 bytes
        g1[1] = (int)((6144u & 0xffffu) << 16);       // tensor_dim0[15:0] = 6144
        g1[2] = (int)(1u << 16);                      // tensor_dim0[31:16]=0 | tensor_dim1=1
        g1[3] = (int)(6144u << 16);                   // tile_dim0 = 6144
        g1[4] = 1;                                    // tile_dim1 = 1, tile_dim2 = 0
        g1[5] = 6144;                                 // tensor_dim0_stride[31:0]
        g1[6] = 0;
        g1[7] = 0;
        i32x4 z4 = {0, 0, 0, 0};                      // groups 2/3 unused (<=2D)
        i32x8 z8 = {0, 0, 0, 0, 0, 0, 0, 0};          // extra group (6-arg toolchain form)
        __builtin_amdgcn_tensor_load_to_lds(g0, g1, z4, z4, z8, 0);
        __builtin_amdgcn_s_wait_tensorcnt(0);         // TENSORcnt -> 0 before barrier
    }
#else
    for (int i = tid; i < NN * 3; i += 256) s_pos[i] = posc[i];
#endif
    __syncthreads();

    const float qx = s_pos[n * 3 + 0], qy = s_pos[n * 3 + 1], qz = s_pos[n * 3 + 2];

    // ---- squared distances, register-resident: thread tid owns j = tid*8 .. tid*8+7
    float d[8];
#pragma unroll
    for (int r = 0; r < 8; ++r) {
        const int j = tid * 8 + r;
        float dx = s_pos[j * 3 + 0] - qx;
        float dy = s_pos[j * 3 + 1] - qy;
        float dz = s_pos[j * 3 + 2] - qz;
        d[r] = dx * dx + dy * dy + dz * dz;
    }

    // ---- 33 sequential argmins; tie-break low index (top_k ordering)
    for (int it = 0; it <= KK; ++it) {
        float best = d[0]; int br = 0;
#pragma unroll
        for (int r = 1; r < 8; ++r)
            if (d[r] < best) { best = d[r]; br = r; }
        int bi = tid * 8 + br;
        // wave32 butterfly reduce (no barriers)
#pragma unroll
        for (int off = 16; off > 0; off >>= 1) {
            float ov = __shfl_xor(best, off, 32);
            int   oi = __shfl_xor(bi,   off, 32);
            if (ov < best || (ov == best && oi < bi)) { best = ov; bi = oi; }
        }
        if (lane == 0) { s_val[w] = best; s_idx[w] = bi; }
        __syncthreads();
        if (tid == 0) {                                // reduce 8 wave results
            float fb = s_val[0]; int fi = s_idx[0];
            for (int q = 1; q < 8; ++q) {
                float ov = s_val[q]; int oi = s_idx[q];
                if (ov < fb || (ov == fb && oi < fi)) { fb = ov; fi = oi; }
            }
            s_sel[it] = fi; s_seld[it] = fb;
        }
        __syncthreads();
        const int winj = s_sel[it];                    // owner invalidates (regs only)
        if ((winj >> 3) == tid) {
#pragma unroll
            for (int r = 0; r < 8; ++r)
                if (r == (winj & 7)) d[r] = FLT_MAX;
        }
    }

    const float mk = mask[b * NN + n];
    const size_t xrow = ((size_t)b * NN + n) * FTOT;

    if (tid < KK) {               // wave 0, lane k = neighbor k (wave32)
        const int k  = tid;
        const int ni = s_sel[k + 1];
        const float dk = s_seld[k + 1];
        float dx = s_pos[ni * 3 + 0] - qx;
        float dy = s_pos[ni * 3 + 1] - qy;
        float dz = s_pos[ni * 3 + 2] - qz;
        float nr = fmaxf(sqrtf(dx * dx + dy * dy + dz * dz), 1e-12f);
        float inv = 1.0f / nr;
        dx *= inv; dy *= inv; dz *= inv;
        // broadcast nearest direction from lane 0 (wave32 shuffle)
        float n0x = __shfl(dx, 0, 32);
        float n0y = __shfl(dy, 0, 32);
        float n0z = __shfl(dz, 0, 32);
        float ct = (k == 0) ? 1.0f : (dx * n0x + dy * n0y + dz * n0z);
        // radius gate: relu(d*rw1) -> dot rw2 -> relu -> sigmoid(mask*.)
        float acc = 0.0f;
        for (int m = 0; m < MM; ++m) {
            float a = dk * rw1[m];
            a = a > 0.0f ? a : 0.0f;
            acc += a * rw2[m];
        }
        acc = acc > 0.0f ? acc : 0.0f;
        float gate = 1.0f / (1.0f + __expf(-acc * mk));
        s_gate[k] = gate;
        for (int j = 0; j < DD; ++j)
            s_gf[k * DD + j] = gate * feac[(size_t)ni * DD + j] * mk;
        X[xrow + k * CC + c] = ct * mk;            // theta, f = k*C + c
    }
    __syncthreads();

    if (tid < DD) {               // k-summed gated features (128..255 block)
        const int j = tid;
        float s = 0.0f, gs = 0.0f;
        for (int k = 0; k < KK; ++k) { s += s_gf[k * DD + j]; gs += s_gate[k]; }
        X[xrow + 128 + c * (2 * DD) + j]      = gs * feac[(size_t)n * DD + j] * mk;  // node_rep half
        X[xrow + 128 + c * (2 * DD) + DD + j] = s;                                   // neighbor half
    }
}

// ---------------------------------------------------------------------------
// Kernel B: fused GEMM  (8192 x 256) @ (256 x 64)  via V_WMMA_F32_16X16X4_F32,
// epilogue: leaky_relu(0.01) * mask, transpose-store to out[b][m][n].
// One wave per 16x16 tile; 8 waves/block -> 2 row-tiles x 4 col-tiles.
// ---------------------------------------------------------------------------
__global__ __launch_bounds__(256) void gemm_wmma_kernel(
    const float* __restrict__ X, const float* __restrict__ W,
    const float* __restrict__ mask, float* __restrict__ out) {

    const int tid  = threadIdx.x;
    const int wave = tid >> 5;
    const int lane = tid & 31;
    const int row_tile = blockIdx.x * 2 + (wave >> 2);  // [0,512)
    const int col_tile = wave & 3;                      // [0,4)
    const int rbase = row_tile * 16;
    const int cbase = col_tile * 16;

    const int lrow  = lane & 15;
    const int khalf = (lane >> 4) << 1;   // lanes 0-15 -> K{0,1}; lanes 16-31 -> K{2,3}

    v8f acc = {};
    const float* ap = X + (size_t)(rbase + lrow) * FTOT + khalf;       // A: row=lrow
    const float* bp = W + (size_t)khalf * MM + cbase + lrow;           // B: col=lrow

    for (int k0 = 0; k0 < FTOT; k0 += 4) {
        v2f a;  a.x = ap[k0];                 a.y = ap[k0 + 1];
        v2f bv; bv.x = bp[(size_t)k0 * MM];   bv.y = bp[(size_t)(k0 + 1) * MM];
        acc = __builtin_amdgcn_wmma_f32_16x16x4_f32(
            /*neg_a=*/false, a, /*neg_b=*/false, bv,
            /*c_mod=*/(short)0, acc, /*reuse_a=*/false, /*reuse_b=*/false);
    }

#pragma unroll
    for (int r = 0; r < 8; ++r) {
        const int row = rbase + r + ((lane >> 4) << 3);  // M index -> flattened (b,n)
        const int m   = cbase + lrow;                    // N index -> output channel
        const int b   = row >> 11;
        const int n   = row & (NN - 1);
        float v = acc[r];
        v = v > 0.0f ? v : 0.01f * v;                    // leaky_relu
        v *= mask[b * NN + n];
        out[((size_t)b * MM + m) * NN + n] = v;
    }
}

// ---------------------------------------------------------------------------
extern "C" void kernel_launch(void* const* d_in, const int* in_sizes, int n_in,
                              void* d_out, int out_size, void* d_ws, size_t ws_size,
                              hipStream_t stream) {
    const float* pos  = (const float*)d_in[0];
    const float* fea  = (const float*)d_in[1];
    const float* mask = (const float*)d_in[2];
    const float* aw   = (const float*)d_in[3];
    const float* sw   = (const float*)d_in[4];
    const float* rw1  = (const float*)d_in[5];
    const float* rw2  = (const float*)d_in[6];
    float* out = (float*)d_out;

    float* X = (float*)d_ws;                       // 4*2048*256 floats = 8 MB
    float* W = X + (size_t)BSZ * NN * FTOT;        // 256*64 floats = 64 KB

    build_w_kernel<<<(FTOT * MM) / 256, 256, 0, stream>>>(aw, sw, W);

    dim3 ga(NN, CC, BSZ);
    knn_gate_kernel<<<ga, 256, 0, stream>>>(pos, fea, mask, rw1, rw2, X);

    // 8192 rows / 16 = 512 row tiles, 2 per block
    gemm_wmma_kernel<<<256, 256, 0, stream>>>(X, W, mask, out);
}